// HistogramQuant_11862699671603
// MI455X (gfx1250) — compile-verified
//
#include <hip/hip_runtime.h>
#include <stdint.h>

#define BINS 2048
#define UPSAMPLE 128
#define NBLOCKS 2048
#define NTHREADS 256

// First RES4 float4-elements (160 MB) are kept L2-resident across both passes;
// the remainder (96 MB) is loaded with TH=NT so it never evicts the resident set.
#define RES4 (160u * 1024u * 1024u / 16u)   // 10,485,760 float4s

typedef float vf4 __attribute__((ext_vector_type(4)));

// ---------------- order-preserving float<->uint keys (global atomic min/max) ------------
__device__ __forceinline__ uint32_t fkey(float f) {
  uint32_t u = __float_as_uint(f);
  return (u & 0x80000000u) ? ~u : (u | 0x80000000u);
}
__device__ __forceinline__ float keyf(uint32_t k) {
  uint32_t u = (k & 0x80000000u) ? (k & 0x7fffffffu) : ~k;
  return __uint_as_float(u);
}

// RT load for the L2-resident prefix, NT (non-temporal) load for the streaming tail.
__device__ __forceinline__ vf4 ldsel(const vf4* __restrict__ p, int i) {
  return (i < (int)RES4) ? p[i] : __builtin_nontemporal_load(&p[i]);
}

// ws layout (uint32 slots):
//  [0] minkey  [1] maxkey
//  [2] cmin(f) [3] cmax(f) [4] hmn(f) [5] hmx(f) [6] scale(f)
//  [7] start_idx(i) [8] dr(i) [9] same flag
//  [16 .. 16+BINS) uint32 counts

// ---------------- K0: per-call reset of workspace accumulators --------------------------
__global__ void k_init(uint32_t* __restrict__ wsu) {
  int i = blockIdx.x * blockDim.x + threadIdx.x;
  if (i == 0) { wsu[0] = 0xffffffffu; wsu[1] = 0u; }
  if (i < BINS) wsu[16 + i] = 0u;
}

// ---------------- K1: global min/max (pass 1 over 256 MB) -------------------------------
__device__ __forceinline__ void mmv(vf4 v, float& mn, float& mx) {
  mn = fminf(mn, fminf(fminf(v.x, v.y), fminf(v.z, v.w)));
  mx = fmaxf(mx, fmaxf(fmaxf(v.x, v.y), fmaxf(v.z, v.w)));
}

__global__ void k_minmax(const vf4* __restrict__ x, int n4, uint32_t* __restrict__ wsu) {
  float mn = 3.402823466e38f, mx = -3.402823466e38f;
  const int stride = gridDim.x * blockDim.x;
  int i = blockIdx.x * blockDim.x + threadIdx.x;
  // unroll x4: 4 independent global_load_b128 in flight per wave
  for (; i + 3 * stride < n4; i += 4 * stride) {
    if (i + 4 * stride < (int)RES4)
      __builtin_prefetch(&x[i + 4 * stride], 0, 0);  // global_prefetch_b8 (RT region only)
    vf4 v0 = ldsel(x, i);
    vf4 v1 = ldsel(x, i + stride);
    vf4 v2 = ldsel(x, i + 2 * stride);
    vf4 v3 = ldsel(x, i + 3 * stride);
    mmv(v0, mn, mx); mmv(v1, mn, mx); mmv(v2, mn, mx); mmv(v3, mn, mx);
  }
  for (; i < n4; i += stride) { vf4 v = ldsel(x, i); mmv(v, mn, mx); }

  __shared__ float smn[NTHREADS], smx[NTHREADS];
  smn[threadIdx.x] = mn; smx[threadIdx.x] = mx;
  __syncthreads();
  for (int off = NTHREADS / 2; off > 0; off >>= 1) {
    if (threadIdx.x < off) {
      smn[threadIdx.x] = fminf(smn[threadIdx.x], smn[threadIdx.x + off]);
      smx[threadIdx.x] = fmaxf(smx[threadIdx.x], smx[threadIdx.x + off]);
    }
    __syncthreads();
  }
  if (threadIdx.x == 0) {
    atomicMin(&wsu[0], fkey(smn[0]));               // global_atomic_min_u32
    atomicMax(&wsu[1], fkey(smx[0]));               // global_atomic_max_u32
  }
}

// ---------------- K2: _adjust_min_max scalar math (1 thread) ----------------------------
__global__ void k_scalars(const float* __restrict__ min_val, const float* __restrict__ max_val,
                          uint32_t* __restrict__ wsu, float* __restrict__ wsf,
                          float* __restrict__ out) {
  float cur_min = keyf(wsu[0]);
  float cur_max = keyf(wsu[1]);
  float mn = min_val[0], mx = max_val[0];
  float cmin  = fminf(cur_min, mn);
  float cmax0 = fmaxf(cur_max, mx);
  float hbw  = (mx - mn) / (float)(BINS * UPSAMPLE);
  float dr_f = ceilf((cmax0 - cmin) / ((float)BINS * hbw));
  float cmax = cmax0 + (dr_f * ((float)BINS * hbw) - (cmax0 - cmin));
  int   start_idx = (int)rintf((mn - cmin) / hbw);   // jnp.round = RNE
  float hmn = truncf(cmin), hmx = truncf(cmax);
  wsf[2] = cmin; wsf[3] = cmax; wsf[4] = hmn; wsf[5] = hmx;
  wsf[6] = (float)BINS / (hmx - hmn);
  wsu[7] = (uint32_t)start_idx;
  wsu[8] = (uint32_t)(int)dr_f;
  wsu[9] = (cmin == mn && cmax == mx) ? 1u : 0u;
  out[BINS]     = cmin;
  out[BINS + 1] = cmax;
}

// ---------------- K3: histogram (pass 2; 160 MB from L2, 96 MB NT from HBM) -------------
__device__ __forceinline__ void binv(vf4 v, float hmn, float hmx, float scale,
                                     uint32_t* __restrict__ lh) {
  float e[4] = {v.x, v.y, v.z, v.w};
#pragma unroll
  for (int k = 0; k < 4; ++k) {
    float xv = e[k];
    if (xv >= hmn && xv <= hmx) {                    // histc drops out-of-range
      int idx = (int)floorf((xv - hmn) * scale);
      idx = idx < 0 ? 0 : (idx > BINS - 1 ? BINS - 1 : idx);
      atomicAdd(&lh[idx], 1u);                       // ds_add_u32 (no return)
    }
  }
}

__global__ void k_hist(const vf4* __restrict__ x, int n4,
                       const float* __restrict__ wsf, uint32_t* __restrict__ wsu) {
  __shared__ uint32_t lh[BINS];                      // 8 KB of the 320 KB/WGP
  for (int i = threadIdx.x; i < BINS; i += blockDim.x) lh[i] = 0u;
  __syncthreads();
  float hmn = wsf[4], hmx = wsf[5], scale = wsf[6];
  const int stride = gridDim.x * blockDim.x;
  int i = blockIdx.x * blockDim.x + threadIdx.x;
  for (; i + 3 * stride < n4; i += 4 * stride) {
    if (i + 4 * stride < (int)RES4)
      __builtin_prefetch(&x[i + 4 * stride], 0, 0);  // global_prefetch_b8 (RT region only)
    vf4 v0 = ldsel(x, i);
    vf4 v1 = ldsel(x, i + stride);
    vf4 v2 = ldsel(x, i + 2 * stride);
    vf4 v3 = ldsel(x, i + 3 * stride);
    binv(v0, hmn, hmx, scale, lh);
    binv(v1, hmn, hmx, scale, lh);
    binv(v2, hmn, hmx, scale, lh);
    binv(v3, hmn, hmx, scale, lh);
  }
  for (; i < n4; i += stride) { vf4 v = ldsel(x, i); binv(v, hmn, hmx, scale, lh); }
  __syncthreads();
  uint32_t* counts = wsu + 16;
  for (int i2 = threadIdx.x; i2 < BINS; i2 += blockDim.x) {
    uint32_t c = lh[i2];
    if (c) atomicAdd(&counts[i2], c);                // global_atomic_add_u32
  }
}

// ---------------- K4: combine histograms analytically (1 block) -------------------------
__device__ __forceinline__ float integral_at(int i, int dr, int start_idx,
                                             const float* spref, const float* sh) {
  int p  = (i + 1) * dr - 1;
  int q  = p - start_idx;
  int qc = q < 0 ? 0 : (q > BINS * UPSAMPLE - 1 ? BINS * UPSAMPLE - 1 : q);
  int j  = qc / UPSAMPLE;
  float frac = (float)(qc % UPSAMPLE);
  return (q < 0) ? 0.0f : spref[j] + (frac + 1.0f) * sh[j];
}

__global__ void k_finalize(const float* __restrict__ hist,
                           const uint32_t* __restrict__ wsu,
                           float* __restrict__ out) {
  __shared__ float sh[BINS];      // old histogram
  __shared__ float spref[BINS];   // exclusive cumsum * UPSAMPLE
  __shared__ float schunk[NTHREADS];
  const int C = BINS / NTHREADS;  // 8 bins per thread
  int t = threadIdx.x;
  float vals[C];
  float s = 0.0f;
#pragma unroll
  for (int k = 0; k < C; ++k) {
    vals[k] = hist[t * C + k];
    sh[t * C + k] = vals[k];
    s += vals[k];
  }
  schunk[t] = s;
  __syncthreads();
  if (t == 0) {                    // serial scan of 256 partials (cheap)
    float acc = 0.0f;
    for (int i = 0; i < NTHREADS; ++i) { float tmp = schunk[i]; schunk[i] = acc; acc += tmp; }
  }
  __syncthreads();
  float acc = schunk[t];
#pragma unroll
  for (int k = 0; k < C; ++k) { spref[t * C + k] = acc * (float)UPSAMPLE; acc += vals[k]; }
  __syncthreads();

  int start_idx = (int)wsu[7];
  int dr        = (int)wsu[8];
  uint32_t same = wsu[9];
  const uint32_t* counts = wsu + 16;
#pragma unroll
  for (int k = 0; k < C; ++k) {
    int i = t * C + k;
    float integ   = integral_at(i, dr, start_idx, spref, sh);
    float shifted = (i == 0) ? 0.0f : integral_at(i - 1, dr, start_idx, spref, sh);
    float interp  = (integ - shifted) / (float)UPSAMPLE;
    float oldpart = same ? sh[i] : interp;
    out[i] = (float)counts[i] + oldpart;
  }
}

// ---------------- launcher --------------------------------------------------------------
extern "C" void kernel_launch(void* const* d_in, const int* in_sizes, int n_in,
                              void* d_out, int out_size, void* d_ws, size_t ws_size,
                              hipStream_t stream) {
  const float* x       = (const float*)d_in[0];
  const float* hist    = (const float*)d_in[1];
  const float* min_val = (const float*)d_in[2];
  const float* max_val = (const float*)d_in[3];
  float*    out = (float*)d_out;
  uint32_t* wsu = (uint32_t*)d_ws;
  float*    wsf = (float*)d_ws;

  int n  = in_sizes[0];       // 64*1024*1024, divisible by 4
  int n4 = n / 4;

  k_init    <<<(BINS + NTHREADS - 1) / NTHREADS, NTHREADS, 0, stream>>>(wsu);
  k_minmax  <<<NBLOCKS, NTHREADS, 0, stream>>>((const vf4*)x, n4, wsu);
  k_scalars <<<1, 1, 0, stream>>>(min_val, max_val, wsu, wsf, out);
  k_hist    <<<NBLOCKS, NTHREADS, 0, stream>>>((const vf4*)x, n4, wsf, wsu);
  k_finalize<<<1, NTHREADS, 0, stream>>>(hist, wsu, out);
}